// SAGE_69587060130174
// MI455X (gfx1250) — compile-verified
//
#include <hip/hip_runtime.h>

typedef __attribute__((ext_vector_type(2))) float v2f;
typedef __attribute__((ext_vector_type(8))) float v8f;

#define N_NODES  50000
#define N_EDGES  800000
#define HID      128
#define OUT_SZ   64

// Monotone float -> uint mapping so unsigned compare == float compare.
__device__ __forceinline__ unsigned f2ord(float f) {
    unsigned u = __float_as_uint(f);
    return (u & 0x80000000u) ? ~u : (u | 0x80000000u);
}

// ---------------------------------------------------------------------------
// out-degree of src, then reciprocal 1/(deg+1e-6)
// ---------------------------------------------------------------------------
__global__ void __launch_bounds__(256) deg_kernel(const int* __restrict__ ei,
                                                  float* __restrict__ deg) {
    int e = blockIdx.x * blockDim.x + threadIdx.x;
    if (e < N_EDGES) atomicAdd(&deg[ei[e]], 1.0f);
}

__global__ void __launch_bounds__(256) rdeg_kernel(float* __restrict__ deg) {
    int i = blockIdx.x * blockDim.x + threadIdx.x;
    if (i < N_NODES) deg[i] = 1.0f / (deg[i] + 1e-6f);
}

// ---------------------------------------------------------------------------
// Wt[c][r] = W[r][c]  (tiny weight matrices, <=64KB)
// ---------------------------------------------------------------------------
__global__ void __launch_bounds__(256) transpose_kernel(
        const float* __restrict__ W, float* __restrict__ Wt, int rows, int cols) {
    int idx = blockIdx.x * blockDim.x + threadIdx.x;
    if (idx < rows * cols) {
        int r = idx / cols, c = idx % cols;
        Wt[c * rows + r] = W[idx];
    }
}

// ---------------------------------------------------------------------------
// C[M x N] = A[M x 128] * W[128 x N] + bias[N]   (fp32 WMMA 16x16x4)
// Wt is W transposed: Wt[N][128]. One wave = 1 N-tile x MT M-tiles.
// blockDim.x = (N/16)*32; gridDim.x = M / (16*MT).
// kh (lane-dependent K offset) folded into base pointers so every load in the
// unrolled K-loop is base + compile-time immediate offset (no address VALU).
// ---------------------------------------------------------------------------
template <int N, int MT>
__global__ void __launch_bounds__(256) wmma_gemm_bias(
        const float* __restrict__ A, const float* __restrict__ Wt,
        const float* __restrict__ bias, float* __restrict__ C) {
    const int lane = threadIdx.x & 31;
    const int wv   = threadIdx.x >> 5;
    const int m0   = blockIdx.x * (16 * MT);
    const int n0   = wv << 4;
    const int l15  = lane & 15;
    const int hi   = lane >> 4;     // 0 or 1
    const int kh   = hi << 1;       // 0 or 2

    const float* brow = Wt + (size_t)(n0 + l15) * HID + kh;
    const float* arow[MT];
#pragma unroll
    for (int t = 0; t < MT; ++t)
        arow[t] = A + (size_t)(m0 + 16 * t + l15) * HID + kh;

    v8f c[MT] = {};
#pragma unroll
    for (int k0 = 0; k0 < HID; k0 += 4) {
        const v2f b = *(const v2f*)(brow + k0);          // 1x global_load_b64
#pragma unroll
        for (int t = 0; t < MT; ++t) {
            const v2f a = *(const v2f*)(arow[t] + k0);   // global_load_b64, imm offset
            c[t] = __builtin_amdgcn_wmma_f32_16x16x4_f32(false, a, false, b,
                                                         (short)0, c[t], false, false);
        }
    }
    const float bv = bias[n0 + l15];
#pragma unroll
    for (int t = 0; t < MT; ++t) {
        float* crow = C + (size_t)(m0 + 16 * t + hi * 8) * N + n0 + l15;
#pragma unroll
        for (int r = 0; r < 8; ++r) crow[r * N] = c[t][r] + bv;
    }
}

// ---------------------------------------------------------------------------
// h = hs @ Wself + (agg * rdeg[row]) @ Wneigh     (N = K = 128)
// ---------------------------------------------------------------------------
template <int MT>
__global__ void __launch_bounds__(256) wmma_gemm_dual(
        const float* __restrict__ Hs,  const float* __restrict__ WselfT,
        const float* __restrict__ Agg, const float* __restrict__ rdeg,
        const float* __restrict__ WneighT, float* __restrict__ C) {
    const int lane = threadIdx.x & 31;
    const int wv   = threadIdx.x >> 5;
    const int m0   = blockIdx.x * (16 * MT);
    const int n0   = wv << 4;
    const int l15  = lane & 15;
    const int hi   = lane >> 4;
    const int kh   = hi << 1;

    const float* bsrow = WselfT  + (size_t)(n0 + l15) * HID + kh;
    const float* bnrow = WneighT + (size_t)(n0 + l15) * HID + kh;
    const float* arow[MT];
    const float* grow[MT];
    float scale[MT];
#pragma unroll
    for (int t = 0; t < MT; ++t) {
        const int m = m0 + 16 * t + l15;
        arow[t]  = Hs  + (size_t)m * HID + kh;
        grow[t]  = Agg + (size_t)m * HID + kh;
        scale[t] = rdeg[m];
    }

    v8f c[MT] = {};
#pragma unroll
    for (int k0 = 0; k0 < HID; k0 += 4) {
        const v2f b = *(const v2f*)(bsrow + k0);
#pragma unroll
        for (int t = 0; t < MT; ++t) {
            const v2f a = *(const v2f*)(arow[t] + k0);
            c[t] = __builtin_amdgcn_wmma_f32_16x16x4_f32(false, a, false, b,
                                                         (short)0, c[t], false, false);
        }
    }
#pragma unroll
    for (int k0 = 0; k0 < HID; k0 += 4) {
        const v2f b = *(const v2f*)(bnrow + k0);
#pragma unroll
        for (int t = 0; t < MT; ++t) {
            const v2f g = *(const v2f*)(grow[t] + k0);
            v2f a; a.x = g.x * scale[t]; a.y = g.y * scale[t];
            c[t] = __builtin_amdgcn_wmma_f32_16x16x4_f32(false, a, false, b,
                                                         (short)0, c[t], false, false);
        }
    }
#pragma unroll
    for (int t = 0; t < MT; ++t) {
        float* crow = C + (size_t)(m0 + 16 * t + hi * 8) * HID + n0 + l15;
#pragma unroll
        for (int r = 0; r < 8; ++r) crow[r * HID] = c[t][r];
    }
}

// ---------------------------------------------------------------------------
// MaxK: keep top-k of 128 per row, zero the rest. One wave32 per row;
// exact k-th largest via 32-step radix select with wave ballots.
// ---------------------------------------------------------------------------
__global__ void __launch_bounds__(256) maxk_kernel(
        const float* __restrict__ H, float* __restrict__ Hs,
        const int* __restrict__ kptr) {
    const int lane = threadIdx.x & 31;
    const int row  = blockIdx.x * 8 + (threadIdx.x >> 5);
    const int k    = kptr[0];

    const float4 v = *(const float4*)(H + (size_t)row * HID + lane * 4);
    const unsigned u0 = f2ord(v.x), u1 = f2ord(v.y),
                   u2 = f2ord(v.z), u3 = f2ord(v.w);

    unsigned thr = 0u;   // max T with count(u >= T) >= k  ==  k-th largest
    for (int bit = 31; bit >= 0; --bit) {
        const unsigned test = thr | (1u << bit);
        int cnt = __popcll(__ballot(u0 >= test)) + __popcll(__ballot(u1 >= test))
                + __popcll(__ballot(u2 >= test)) + __popcll(__ballot(u3 >= test));
        if (cnt >= k) thr = test;   // uniform branch across the wave
    }
    float4 o;
    o.x = (u0 >= thr) ? v.x : 0.0f;
    o.y = (u1 >= thr) ? v.y : 0.0f;
    o.z = (u2 >= thr) ? v.z : 0.0f;
    o.w = (u3 >= thr) ? v.w : 0.0f;
    *(float4*)(Hs + (size_t)row * HID + lane * 4) = o;
}

// ---------------------------------------------------------------------------
// agg[src[e]] += hs[dst[e]]  — wave per edge, 4 floats per lane.
// hs is 75% zeros after MaxK: skip zero atomics (4x less L2 atomic traffic).
// ---------------------------------------------------------------------------
__global__ void __launch_bounds__(256) scatter_kernel(
        const float* __restrict__ Hs, const int* __restrict__ ei,
        float* __restrict__ Agg) {
    const int lane = threadIdx.x & 31;
    const int e    = blockIdx.x * 8 + (threadIdx.x >> 5);
    const int s    = ei[e];
    const int d    = ei[N_EDGES + e];
    const float4 v = *(const float4*)(Hs + (size_t)d * HID + lane * 4);
    float* out = Agg + (size_t)s * HID + lane * 4;
    if (v.x != 0.0f) atomicAdd(out + 0, v.x);
    if (v.y != 0.0f) atomicAdd(out + 1, v.y);
    if (v.z != 0.0f) atomicAdd(out + 2, v.z);
    if (v.w != 0.0f) atomicAdd(out + 3, v.w);
}

// ---------------------------------------------------------------------------
extern "C" void kernel_launch(void* const* d_in, const int* in_sizes, int n_in,
                              void* d_out, int out_size, void* d_ws, size_t ws_size,
                              hipStream_t stream) {
    const float* x      = (const float*)d_in[0];
    const float* W_in   = (const float*)d_in[1];
    const float* b_in   = (const float*)d_in[2];
    const float* W_self = (const float*)d_in[3];
    const float* W_ngh  = (const float*)d_in[4];
    const float* W_out  = (const float*)d_in[5];
    const float* b_out  = (const float*)d_in[6];
    const int*   ei     = (const int*)d_in[7];
    const int*   kptr   = (const int*)d_in[8];
    float*       out    = (float*)d_out;

    const size_t NH = (size_t)N_NODES * HID;
    const size_t WW = (size_t)HID * HID;          // 16384
    float* h      = (float*)d_ws;
    float* hs     = h + NH;
    float* agg    = hs + NH;
    float* rdeg   = agg + NH;
    float* WinT   = rdeg + 50048;                 // keep 16B alignment
    float* WselfT = WinT + WW;                    // 2 layers
    float* WnghT  = WselfT + 2 * WW;              // 2 layers
    float* WoutT  = WnghT + 2 * WW;               // 64 x 128

    // transpose all weights once per call (tiny, L2-resident afterwards)
    transpose_kernel<<<(int)(WW / 256), 256, 0, stream>>>(W_in, WinT, HID, HID);
    for (int l = 0; l < 2; ++l) {
        transpose_kernel<<<(int)(WW / 256), 256, 0, stream>>>(
            W_self + l * WW, WselfT + l * WW, HID, HID);
        transpose_kernel<<<(int)(WW / 256), 256, 0, stream>>>(
            W_ngh + l * WW, WnghT + l * WW, HID, HID);
    }
    transpose_kernel<<<(int)(HID * OUT_SZ / 256), 256, 0, stream>>>(
        W_out, WoutT, HID, OUT_SZ);

    // degree + reciprocal (recomputed every call -> deterministic)
    hipMemsetAsync(rdeg, 0, N_NODES * sizeof(float), stream);
    deg_kernel<<<(N_EDGES + 255) / 256, 256, 0, stream>>>(ei, rdeg);
    rdeg_kernel<<<(N_NODES + 255) / 256, 256, 0, stream>>>(rdeg);

    // h = x @ W_in + b_in     (3125 M-tiles = 625 blocks x 5 tiles/wave)
    wmma_gemm_bias<HID, 5><<<625, 256, 0, stream>>>(x, WinT, b_in, h);

    for (int layer = 0; layer < 2; ++layer) {
        maxk_kernel<<<N_NODES / 8, 256, 0, stream>>>(h, hs, kptr);
        hipMemsetAsync(agg, 0, NH * sizeof(float), stream);
        scatter_kernel<<<N_EDGES / 8, 256, 0, stream>>>(hs, ei, agg);
        wmma_gemm_dual<5><<<625, 256, 0, stream>>>(
            hs, WselfT + (size_t)layer * WW, agg, rdeg,
            WnghT + (size_t)layer * WW, h);
    }

    // out = h @ W_out + b_out   (N = 64 -> 4 waves / block)
    wmma_gemm_bias<OUT_SZ, 5><<<625, (OUT_SZ / 16) * 32, 0, stream>>>(
        h, WoutT, b_out, out);
}